// WeightedDecompositionKernel_18683107737744
// MI455X (gfx1250) — compile-verified
//
#include <hip/hip_runtime.h>
#include <hip/hip_bf16.h>
#include <math.h>

typedef __attribute__((ext_vector_type(2))) float v2f;
typedef __attribute__((ext_vector_type(8))) float v8f;

#define L_SEQ 512
#define NSYM  30
#define DEMB  64
#define DEG   8
#define NROW1 128
#define NROW2 128

// ---------------------------------------------------------------------------
// Kernel A: S = A * A^T  (30x30, zero-padded to 32x32) via v_wmma_f32_16x16x4_f32
// A is staged once into a zero-padded 32x64 LDS buffer so the WMMA loop reads
// operands with unconditional ds_load_b64 (kk is always even -> 8B aligned),
// avoiding per-element EXEC-masked global loads.
// One workgroup, 4 waves; wave w owns 16x16 tile (ma, nb).
// A-operand layout (32-bit A 16x4): lane M = lane&15; VGPR0 = K0 / K2,
// VGPR1 = K1 / K3 split across lane halves. Since B = A^T, the B-operand
// addressing reduces to the identical expression on the column row index.
// ---------------------------------------------------------------------------
__global__ __launch_bounds__(128) void s_gemm_kernel(const float* __restrict__ A,
                                                     float* __restrict__ S32)
{
    __shared__ float Ash[32 * DEMB];      // 8 KB, rows 30/31 zero

    const int tid   = threadIdx.x;
    const int wave  = tid >> 5;
    const int lane  = tid & 31;
    const int r     = lane & 15;
    const int khalf = lane >> 4;          // 0: K{0,1}, 1: K{2,3}
    const int ma    = (wave >> 1) << 4;
    const int nb    = (wave & 1) << 4;

    for (int idx = tid; idx < 32 * DEMB; idx += 128)
        Ash[idx] = (idx < NSYM * DEMB) ? A[idx] : 0.0f;
    __syncthreads();

    const float* Arow = Ash + (ma + r) * DEMB;
    const float* Brow = Ash + (nb + r) * DEMB;

    v8f acc = {};
#pragma unroll
    for (int k0 = 0; k0 < DEMB; k0 += 4) {
        const int kk = k0 + khalf * 2;
        const v2f a = *(const v2f*)(Arow + kk);
        const v2f b = *(const v2f*)(Brow + kk);
        acc = __builtin_amdgcn_wmma_f32_16x16x4_f32(false, a, false, b,
                                                    (short)0, acc, false, false);
    }

    // C/D layout: N = lane&15, M = v + 8*(lane>>4)
#pragma unroll
    for (int v = 0; v < 8; ++v) {
        const int M = v + khalf * 8;
        S32[(ma + M) * 32 + (nb + r)] = acc[v];
    }
}

// ---------------------------------------------------------------------------
// Kernel B: self terms.  rnorm[b] = rsqrt( wdk(diag(S)[X_b]) )
// One 128-thread block per sequence (b<128 -> X1 row, else X2 row).
// ---------------------------------------------------------------------------
__global__ __launch_bounds__(128) void norm_kernel(const int* __restrict__ X1,
                                                   const int* __restrict__ X2,
                                                   const int* __restrict__ graph,
                                                   const float* __restrict__ S32,
                                                   float* __restrict__ rnorm)
{
    __shared__ float dbuf[L_SEQ];
    __shared__ int   gsh[L_SEQ * DEG];
    __shared__ float red[128];

    const int b   = blockIdx.x;
    const int tid = threadIdx.x;
    const int* row = (b < NROW1) ? (X1 + b * L_SEQ) : (X2 + (b - NROW1) * L_SEQ);

    for (int idx = tid; idx < L_SEQ * DEG; idx += 128) gsh[idx] = graph[idx];
    for (int l = tid; l < L_SEQ; l += 128) {
        const int s = row[l];
        dbuf[l] = S32[s * 33];            // diag of padded 32x32 S
    }
    __syncthreads();

    float part = 0.0f;
    for (int l = tid; l < L_SEQ; l += 128) {
        float nsum = 0.0f;
#pragma unroll
        for (int k = 0; k < DEG; ++k) {
            const unsigned gv = (unsigned)gsh[l * DEG + k];
            nsum += (gv < (unsigned)L_SEQ) ? dbuf[gv] : 0.0f;   // idx L -> zero pad
        }
        part += dbuf[l] * nsum;
    }
    red[tid] = part;
    __syncthreads();
    for (int s = 64; s > 0; s >>= 1) {
        if (tid < s) red[tid] += red[tid + s];
        __syncthreads();
    }
    if (tid == 0) rnorm[b] = rsqrtf(red[0]);
}

// ---------------------------------------------------------------------------
// Kernel C: main cross kernel.  Block = (i, jblk): 8 waves, one (i,j) pair per
// wave32.  S table + graph + X1 row cached in LDS; per-wave c-buffer in LDS;
// conflict-free neighbor gathers; __shfl_xor wave reduction.
// ---------------------------------------------------------------------------
__global__ __launch_bounds__(256) void cross_kernel(const int* __restrict__ X1,
                                                    const int* __restrict__ X2,
                                                    const int* __restrict__ graph,
                                                    const float* __restrict__ S32,
                                                    const float* __restrict__ rnorm,
                                                    const float* __restrict__ a_p,
                                                    const float* __restrict__ gamma_p,
                                                    float* __restrict__ out)
{
    __shared__ float Ssh[32 * 32];          // 4 KB
    __shared__ int   x1sh[L_SEQ];           // 2 KB
    __shared__ int   gsh[L_SEQ * DEG];      // 16 KB
    __shared__ float cbuf[8][L_SEQ];        // 16 KB

    const int i    = blockIdx.y;
    const int jblk = blockIdx.x;
    const int tid  = threadIdx.x;
    const int wave = tid >> 5;
    const int lane = tid & 31;
    const int j    = jblk * 8 + wave;

    for (int idx = tid; idx < 1024; idx += 256)         Ssh[idx]  = S32[idx];
    for (int l = tid; l < L_SEQ; l += 256)              x1sh[l]   = X1[i * L_SEQ + l];
    for (int idx = tid; idx < L_SEQ * DEG; idx += 256)  gsh[idx]  = graph[idx];
    __syncthreads();

    // c_l = S[x1_l, x2_l]
    const int* x2row = X2 + j * L_SEQ;
    for (int l = lane; l < L_SEQ; l += 32) {
        const int s = x1sh[l];
        const int t = x2row[l];
        cbuf[wave][l] = Ssh[s * 32 + t];
    }
    __syncthreads();

    // K = sum_l c_l * sum_{k in graph(l)} c_{g}
    float part = 0.0f;
    for (int l = lane; l < L_SEQ; l += 32) {
        float nsum = 0.0f;
#pragma unroll
        for (int k = 0; k < DEG; ++k) {
            const unsigned gv = (unsigned)gsh[l * DEG + k];
            nsum += (gv < (unsigned)L_SEQ) ? cbuf[wave][gv] : 0.0f;
        }
        part += cbuf[wave][l] * nsum;
    }

    // wave32 reduction
#pragma unroll
    for (int off = 16; off > 0; off >>= 1)
        part += __shfl_xor(part, off, 32);

    if (lane == 0) {
        const float K  = part * rnorm[i] * rnorm[NROW1 + j];
        const float av = a_p[0];
        const float gv = gamma_p[0];
        out[i * NROW2 + j] = av * av * powf(K, gv);
    }
}

// ---------------------------------------------------------------------------
extern "C" void kernel_launch(void* const* d_in, const int* in_sizes, int n_in,
                              void* d_out, int out_size, void* d_ws, size_t ws_size,
                              hipStream_t stream)
{
    const int*   X1    = (const int*)d_in[0];
    const int*   X2    = (const int*)d_in[1];
    const int*   graph = (const int*)d_in[2];
    const float* A     = (const float*)d_in[3];
    const float* a_p   = (const float*)d_in[4];
    const float* g_p   = (const float*)d_in[5];

    float* S32   = (float*)d_ws;     // 1024 floats: padded 32x32 Gram matrix
    float* rnorm = S32 + 1024;       // 256 floats: 1/sqrt(k1) then 1/sqrt(k2)
    float* out   = (float*)d_out;

    s_gemm_kernel<<<1, 128, 0, stream>>>(A, S32);
    norm_kernel<<<NROW1 + NROW2, 128, 0, stream>>>(X1, X2, graph, S32, rnorm);
    cross_kernel<<<dim3(16, NROW1), 256, 0, stream>>>(X1, X2, graph, S32, rnorm,
                                                      a_p, g_p, out);
}